// GCN_Fusion_surface_21680994910211
// MI455X (gfx1250) — compile-verified
//
#include <hip/hip_runtime.h>
#include <hip/hip_bf16.h>

typedef __attribute__((ext_vector_type(2))) float v2f;
typedef __attribute__((ext_vector_type(8))) float v8f;

#define NBATCH 2
#define NVERT  2048
#define NCH    64
#define NS     4
#define KLOC   10
#define KGLB   100
#define ROWS   (NBATCH * NVERT)   // 4096
#define FOUTC  ((NS + 1) * NCH)   // 320

// ---------------------------------------------------------------------------
// Kernel 1: kNN. One 256-thread block per (b,v) row. Candidate keys
// (distbits<<32 | index) live in registers; per iteration: unrolled register
// min, wave32 shfl_xor min-reduce, one shared atomicMin per wave.
// ---------------------------------------------------------------------------
__global__ void knn_kernel(const float* __restrict__ verts,
                           int* __restrict__ idx_g) {
  __shared__ unsigned long long best;
  __shared__ int winner;
  const int row  = blockIdx.x;            // b*NVERT + i
  const int b    = row >> 11;
  const int i    = row & (NVERT - 1);
  const int lane = threadIdx.x & 31;
  const float px = verts[row * 3 + 0];
  const float py = verts[row * 3 + 1];
  const float pz = verts[row * 3 + 2];

  unsigned long long keys[8];
#pragma unroll
  for (int e = 0; e < 8; ++e) {
    const int j  = threadIdx.x + (e << 8);
    const int gr = b * NVERT + j;
    float dx = verts[gr * 3 + 0] - px;
    float dy = verts[gr * 3 + 1] - py;
    float dz = verts[gr * 3 + 2] - pz;
    float d  = dx * dx + dy * dy + dz * dz;
    if (j == i) d = __builtin_inff();
    keys[e] = ((unsigned long long)__float_as_uint(d) << 32) | (unsigned)j;
  }
  if (threadIdx.x == 0) best = ~0ull;
  __syncthreads();

  for (int n = 0; n < KGLB; ++n) {
    unsigned long long key = keys[0];
#pragma unroll
    for (int e = 1; e < 8; ++e) key = (keys[e] < key) ? keys[e] : key;
#pragma unroll
    for (int o = 16; o > 0; o >>= 1) {
      unsigned long long other = __shfl_xor(key, o, 32);
      key = (other < key) ? other : key;
    }
    if (lane == 0) atomicMin(&best, key);
    __syncthreads();                       // all atomics done
    if (threadIdx.x == 0) {
      int w = (int)(best & 0xffffffffu);
      winner = w;
      idx_g[row * KGLB + n] = w;
      best = ~0ull;
    }
    __syncthreads();                       // winner visible, best reset
    const int w = winner;
    if ((w & 255) == threadIdx.x) {
      const int slot = w >> 8;
#pragma unroll
      for (int e = 0; e < 8; ++e)
        if (slot == e) keys[e] = ~0ull;
    }
  }
}

// ---------------------------------------------------------------------------
// Helper: normalized direction for neighbor n of `row`, written into LDS in
// WMMA-A-fragment layout: afr[tile][m]      = (x, y)   (lanes 0-15 -> K=0,1)
//                         afr[tile][16 + m] = (z, 0)   (lanes 16-31 -> K=2,3)
// Inner loops then read ONE branchless ds_load_b64 per WMMA: afr[q][lane].
// ---------------------------------------------------------------------------

// Kernel 2: conv_surface. One wave per (b,v). V_WMMA_F32_16X16X4_F32.
// No validity mask needed: padded (zero) dn rows give raw theta 0, and
// relu(max_k raw) == max_k relu, which is >= 0 regardless.
__global__ void conv_surface_kernel(const float* __restrict__ verts,
                                    const int* __restrict__ idxg,
                                    int k, int ntiles,
                                    const float* __restrict__ dirs,   // (3,256)
                                    float* __restrict__ outraw) {     // (ROWS,64)
  __shared__ v2f   afr[4][7][32];
  __shared__ float osh[4][NCH];
  const int wave = threadIdx.x >> 5;
  const int lane = threadIdx.x & 31;
  const int l15  = lane & 15;
  const int row  = blockIdx.x * 4 + wave;
  const int b    = row >> 11;
  const float px = verts[row * 3 + 0];
  const float py = verts[row * 3 + 1];
  const float pz = verts[row * 3 + 2];

  for (int n = lane; n < ntiles * 16; n += 32) {
    float x = 0.f, y = 0.f, z = 0.f;
    if (n < k) {
      int j  = idxg[row * KGLB + n];
      int gr = b * NVERT + j;
      float dx = verts[gr * 3 + 0] - px;
      float dy = verts[gr * 3 + 1] - py;
      float dz = verts[gr * 3 + 2] - pz;
      float inv = 1.f / fmaxf(sqrtf(dx * dx + dy * dy + dz * dz), 1e-12f);
      x = dx * inv; y = dy * inv; z = dz * inv;
    }
    const int q = n >> 4, m = n & 15;
    v2f xy; xy[0] = x; xy[1] = y;
    v2f z0; z0[0] = z; z0[1] = 0.f;
    afr[wave][q][m]      = xy;
    afr[wave][q][16 + m] = z0;
  }
  osh[wave][lane] = 0.f;
  osh[wave][lane + 32] = 0.f;
  __syncthreads();

  v2f bfrag[16];
#pragma unroll
  for (int t = 0; t < 16; ++t) {
    int col = t * 16 + l15;
    float s0 = dirs[col], s1 = dirs[256 + col], s2 = dirs[512 + col];
    float inv = 1.f / fmaxf(sqrtf(s0 * s0 + s1 * s1 + s2 * s2), 1e-12f);
    bfrag[t][0] = ((lane < 16) ? s0 : s2) * inv;
    bfrag[t][1] = (lane < 16) ? (s1 * inv) : 0.f;
  }

  for (int t = 0; t < 16; ++t) {
    float m = 0.f;                          // relu lower bound
    for (int q = 0; q < ntiles; ++q) {
      v2f a = afr[wave][q][lane];           // one ds_load_b64, no divergence
      v8f cz = {};
      v8f d = __builtin_amdgcn_wmma_f32_16x16x4_f32(false, a, false, bfrag[t],
                                                    (short)0, cz, false, false);
#pragma unroll
      for (int r = 0; r < 8; ++r) m = fmaxf(m, d[r]);
    }
    m = fmaxf(m, __shfl_xor(m, 16, 32));
    if (lane < 16) osh[wave][(t & 3) * 16 + lane] += m;   // sum over S
  }
  __syncthreads();
  outraw[row * NCH + lane]      = osh[wave][lane];
  outraw[row * NCH + lane + 32] = osh[wave][lane + 32];
}

// ---------------------------------------------------------------------------
// Kernel 3: fout = fm @ W + bias  (4096x64 @ 64x320). One 16x16 tile per wave,
// 16 accumulating WMMA f32 16x16x4 steps over K=64. Branchless fragment loads:
// A is a single global_load_b64 (two consecutive K values per lane half).
// ---------------------------------------------------------------------------
__global__ void gemm_fout_kernel(const float* __restrict__ fm,    // (ROWS,64)
                                 const float* __restrict__ W,     // (64,320)
                                 const float* __restrict__ bias,  // (320)
                                 float* __restrict__ fout) {      // (ROWS,320)
  const int wave = threadIdx.x >> 5;
  const int lane = threadIdx.x & 31;
  const int l15  = lane & 15;
  const int tile = blockIdx.x * 4 + wave;     // 256*20 = 5120 tiles
  const int tm = tile / 20, tn = tile % 20;
  const int row0 = tm * 16, col0 = tn * 16;
  const int koff = (lane < 16) ? 0 : 2;       // lane half's K sub-offset
  const float* aptr = fm + (row0 + l15) * NCH + koff;
  const float* bptr = W + koff * FOUTC + col0 + l15;
  v8f acc = {};
  for (int kk = 0; kk < NCH; kk += 4) {
    v2f a = *(const v2f*)(aptr + kk);         // K=kk+koff, kk+koff+1
    v2f bb;
    bb[0] = bptr[(kk + 0) * FOUTC];
    bb[1] = bptr[(kk + 1) * FOUTC];
    acc = __builtin_amdgcn_wmma_f32_16x16x4_f32(false, a, false, bb,
                                                (short)0, acc, false, false);
  }
  const float bv = bias[col0 + l15];
  const int moff = (lane < 16) ? 0 : 8;
#pragma unroll
  for (int r = 0; r < 8; ++r)
    fout[(row0 + r + moff) * FOUTC + col0 + l15] = acc[r] + bv;
}

// ---------------------------------------------------------------------------
// Kernel 4: conv_layer: center + sum_s max_k( relu(theta)*support ).
// Tiles 0..5 (n=0..95) fully valid -> unmasked. Tail tile 6: valid rows are
// M=0..3 (lower lane half, r<4) only.
// ---------------------------------------------------------------------------
__global__ void conv_layer_kernel(const float* __restrict__ verts,
                                  const int* __restrict__ idxg,
                                  const float* __restrict__ dirs,   // dirs_g1
                                  const float* __restrict__ fout,   // (ROWS,320)
                                  float* __restrict__ outraw) {     // (ROWS,64)
  __shared__ v2f   afr[4][7][32];
  __shared__ int   nbsh[4][112];
  __shared__ float osh[4][NCH];
  const int wave = threadIdx.x >> 5;
  const int lane = threadIdx.x & 31;
  const int l15  = lane & 15;
  const int row  = blockIdx.x * 4 + wave;
  const int b    = row >> 11;
  const float px = verts[row * 3 + 0];
  const float py = verts[row * 3 + 1];
  const float pz = verts[row * 3 + 2];

  for (int n = lane; n < 112; n += 32) {
    float x = 0.f, y = 0.f, z = 0.f;
    int gr = row;
    if (n < KGLB) {
      int j = idxg[row * KGLB + n];
      gr = b * NVERT + j;
      float dx = verts[gr * 3 + 0] - px;
      float dy = verts[gr * 3 + 1] - py;
      float dz = verts[gr * 3 + 2] - pz;
      float inv = 1.f / fmaxf(sqrtf(dx * dx + dy * dy + dz * dz), 1e-12f);
      x = dx * inv; y = dy * inv; z = dz * inv;
    }
    const int q = n >> 4, m = n & 15;
    v2f xy; xy[0] = x; xy[1] = y;
    v2f z0; z0[0] = z; z0[1] = 0.f;
    afr[wave][q][m]      = xy;
    afr[wave][q][16 + m] = z0;
    nbsh[wave][n] = gr;
  }
  osh[wave][lane]      = fout[row * FOUTC + lane];        // center
  osh[wave][lane + 32] = fout[row * FOUTC + lane + 32];
  __syncthreads();

  v2f bfrag[16];
#pragma unroll
  for (int t = 0; t < 16; ++t) {
    int col = t * 16 + l15;
    float s0 = dirs[col], s1 = dirs[256 + col], s2 = dirs[512 + col];
    float inv = 1.f / fmaxf(sqrtf(s0 * s0 + s1 * s1 + s2 * s2), 1e-12f);
    bfrag[t][0] = ((lane < 16) ? s0 : s2) * inv;
    bfrag[t][1] = (lane < 16) ? (s1 * inv) : 0.f;
  }

  const int moff = (lane < 16) ? 0 : 8;
  for (int t = 0; t < 16; ++t) {
    const float* supcol = fout + NCH + t * 16 + l15;      // + nb*FOUTC below
    float m = -__builtin_inff();
    // Fully-valid neighbor tiles (n = 0..95): no masking.
    for (int q = 0; q < 6; ++q) {
      v2f a = afr[wave][q][lane];           // one ds_load_b64, no divergence
      v8f cz = {};
      v8f d = __builtin_amdgcn_wmma_f32_16x16x4_f32(false, a, false, bfrag[t],
                                                    (short)0, cz, false, false);
#pragma unroll
      for (int r = 0; r < 8; ++r) {
        float sup = supcol[nbsh[wave][q * 16 + r + moff] * FOUTC];
        m = fmaxf(m, fmaxf(d[r], 0.f) * sup);
      }
    }
    // Tail tile (n = 96..99 valid): only lower lane half, r < 4.
    {
      v2f a = afr[wave][6][lane];
      v8f cz = {};
      v8f d = __builtin_amdgcn_wmma_f32_16x16x4_f32(false, a, false, bfrag[t],
                                                    (short)0, cz, false, false);
      if (lane < 16) {
#pragma unroll
        for (int r = 0; r < 4; ++r) {
          float sup = supcol[nbsh[wave][96 + r] * FOUTC];
          m = fmaxf(m, fmaxf(d[r], 0.f) * sup);
        }
      }
    }
    m = fmaxf(m, __shfl_xor(m, 16, 32));
    if (lane < 16) osh[wave][(t & 3) * 16 + lane] += m;
  }
  __syncthreads();
  outraw[row * NCH + lane]      = osh[wave][lane];
  outraw[row * NCH + lane + 32] = osh[wave][lane + 32];
}

// ---------------------------------------------------------------------------
// Batch-norm: per-channel sum/sumsq (single block) + apply relu/affine.
// ---------------------------------------------------------------------------
__global__ void bn_stats_kernel(const float* __restrict__ x,   // (ROWS,64)
                                float* __restrict__ stats) {   // [64 sum | 64 sq]
  __shared__ float ss[NCH], sq[NCH];
  const int tid = threadIdx.x;                 // 1024 threads
  if (tid < NCH) { ss[tid] = 0.f; sq[tid] = 0.f; }
  __syncthreads();
  const int c = tid & 63, p = tid >> 6;        // 16 partitions
  float a = 0.f, b2 = 0.f;
  for (int r = p; r < ROWS; r += 16) {
    float v = x[r * NCH + c];
    a += v; b2 += v * v;
  }
  atomicAdd(&ss[c], a);
  atomicAdd(&sq[c], b2);
  __syncthreads();
  if (tid < NCH) { stats[tid] = ss[tid]; stats[NCH + tid] = sq[tid]; }
}

__global__ void bn_apply_kernel(const float* __restrict__ x,
                                const float* __restrict__ stats,
                                const float* __restrict__ gamma,
                                const float* __restrict__ beta,
                                float* __restrict__ dst,
                                int dst_stride, int dst_off) {
  const int gid = blockIdx.x * blockDim.x + threadIdx.x;   // ROWS*64
  const int r = gid >> 6, c = gid & 63;
  const float inv_n = 1.f / (float)ROWS;
  float mean = stats[c] * inv_n;
  float var  = stats[NCH + c] * inv_n - mean * mean;
  float y = (x[gid] - mean) * rsqrtf(var + 1e-5f) * gamma[c] + beta[c];
  dst[r * dst_stride + dst_off + c] = fmaxf(y, 0.f);
}

// ---------------------------------------------------------------------------
extern "C" void kernel_launch(void* const* d_in, const int* in_sizes, int n_in,
                              void* d_out, int out_size, void* d_ws, size_t ws_size,
                              hipStream_t stream) {
  const float* verts    = (const float*)d_in[0];
  const float* dirs_l   = (const float*)d_in[1];
  const float* dirs_g0  = (const float*)d_in[2];
  const float* dirs_g1  = (const float*)d_in[3];
  const float* Wg1      = (const float*)d_in[4];
  const float* bias_g1  = (const float*)d_in[5];
  const float* gamma_l  = (const float*)d_in[6];
  const float* beta_l   = (const float*)d_in[7];
  const float* gamma_g0 = (const float*)d_in[8];
  const float* beta_g0  = (const float*)d_in[9];
  const float* gamma_g1 = (const float*)d_in[10];
  const float* beta_g1  = (const float*)d_in[11];
  float* out = (float*)d_out;

  char* ws = (char*)d_ws;
  size_t off = 0;
  int*   idx_g     = (int*)(ws + off);   off += (size_t)ROWS * KGLB * 4;
  float* fm_l_raw  = (float*)(ws + off); off += (size_t)ROWS * NCH * 4;
  float* fm_g_raw  = (float*)(ws + off); off += (size_t)ROWS * NCH * 4;
  float* fm_g_bn   = (float*)(ws + off); off += (size_t)ROWS * NCH * 4;
  float* fm_g1_raw = (float*)(ws + off); off += (size_t)ROWS * NCH * 4;
  float* fout      = (float*)(ws + off); off += (size_t)ROWS * FOUTC * 4;
  float* stats     = (float*)(ws + off); off += 2 * NCH * 4;

  // 1) kNN (top-100; first 10 double as local neighborhood)
  knn_kernel<<<ROWS, 256, 0, stream>>>(verts, idx_g);

  // 2) conv_surface local (k=10, 1 neighbor tile) and global (k=100, 7 tiles)
  conv_surface_kernel<<<ROWS / 4, 128, 0, stream>>>(verts, idx_g, KLOC, 1,
                                                    dirs_l, fm_l_raw);
  conv_surface_kernel<<<ROWS / 4, 128, 0, stream>>>(verts, idx_g, KGLB, 7,
                                                    dirs_g0, fm_g_raw);

  // 3) BN(local) -> out[:, :, 0:64]
  bn_stats_kernel<<<1, 1024, 0, stream>>>(fm_l_raw, stats);
  bn_apply_kernel<<<ROWS * NCH / 256, 256, 0, stream>>>(
      fm_l_raw, stats, gamma_l, beta_l, out, 2 * NCH, 0);

  // 4) BN(global layer 0) -> fm_g_bn (workspace)
  bn_stats_kernel<<<1, 1024, 0, stream>>>(fm_g_raw, stats);
  bn_apply_kernel<<<ROWS * NCH / 256, 256, 0, stream>>>(
      fm_g_raw, stats, gamma_g0, beta_g0, fm_g_bn, NCH, 0);

  // 5) fout = fm_g_bn @ W + bias   (WMMA GEMM)
  gemm_fout_kernel<<<(256 * 20) / 4, 128, 0, stream>>>(fm_g_bn, Wg1, bias_g1, fout);

  // 6) conv_layer aggregation -> fm_g1_raw
  conv_layer_kernel<<<ROWS / 4, 128, 0, stream>>>(verts, idx_g, dirs_g1, fout,
                                                  fm_g1_raw);

  // 7) BN(global layer 1) -> out[:, :, 64:128]
  bn_stats_kernel<<<1, 1024, 0, stream>>>(fm_g1_raw, stats);
  bn_apply_kernel<<<ROWS * NCH / 256, 256, 0, stream>>>(
      fm_g1_raw, stats, gamma_g1, beta_g1, out, 2 * NCH, NCH);
}